// CoreProcessor_79740362818145
// MI455X (gfx1250) — compile-verified
//
#include <hip/hip_runtime.h>
#include <math.h>

typedef __attribute__((ext_vector_type(2))) float v2f;
typedef __attribute__((ext_vector_type(8))) float v8f;

#define TOK_PER_WG 16
#define DDIM 256
#define M_MEM 16384
#define KSEL 32
#define NT 512                 // memory rows per score chunk
#define NCHUNK (M_MEM / NT)
#define SC_STRIDE 516          // padded to avoid LDS bank conflicts
#define LN_EPS 1e-5f

__device__ __forceinline__ float wave_sum_f(float v) {
#pragma unroll
  for (int o = 16; o > 0; o >>= 1) v += __shfl_xor(v, o, 32);
  return v;
}
__device__ __forceinline__ float wave_max_f(float v) {
#pragma unroll
  for (int o = 16; o > 0; o >>= 1) v = fmaxf(v, __shfl_xor(v, o, 32));
  return v;
}

__global__ __launch_bounds__(512, 1)
void core_processor_kernel(const float* __restrict__ x,
                           const float* __restrict__ mem_keys,
                           const float* __restrict__ mem_values,
                           const float* __restrict__ W_fuse,
                           const float* __restrict__ b_fuse,
                           const float* __restrict__ ln_g,
                           const float* __restrict__ ln_b,
                           const float* __restrict__ W_op,
                           const float* __restrict__ b_op,
                           float* __restrict__ out)
{
  __shared__ __align__(16) float xs[TOK_PER_WG * DDIM];       // x tile, then fused tile
  __shared__ __align__(16) float sc[TOK_PER_WG * SC_STRIDE];  // score chunk, reused as h tile
  __shared__ float tv[TOK_PER_WG * KSEL];                     // top-k values
  __shared__ int   ti[TOK_PER_WG * KSEL];                     // top-k indices

  const int tid  = threadIdx.x;
  const int lane = tid & 31;
  const int w    = tid >> 5;                 // wave id == local token id (0..15)
  const long tokBase = (long)blockIdx.x * TOK_PER_WG;

  const int rowA  = lane & 15;               // WMMA A: M index held by this lane
  const int khalf = (lane >> 4) << 1;        // WMMA f32 16x16x4: lanes 16-31 hold K=2,3

  // ---------- Phase 1: stage x tile (16 x 256) into LDS ----------
  {
    const float4* src = (const float4*)(x + tokBase * DDIM);
    float4* dst = (float4*)xs;
    dst[tid]       = src[tid];
    dst[tid + 512] = src[tid + 512];
  }
  tv[w * KSEL + lane] = -INFINITY;
  ti[w * KSEL + lane] = 0;
  float curMin = -INFINITY;
  int   minPos = 0;
  __syncthreads();

  // ---------- Phase 2: scores GEMM (WMMA f32) fused with running top-32 ----------
  for (int chunk = 0; chunk < NCHUNK; ++chunk) {
    const int chunkBase = chunk * NT;
    // each wave computes two 16x16 score tiles of this 16 x 512 chunk
#pragma unroll
    for (int t2 = 0; t2 < 2; ++t2) {
      const int tile = w * 2 + t2;                  // 0..31
      const int colB = chunkBase + tile * 16 + (lane & 15);
      const float* ap = xs + rowA * DDIM + khalf;
      const float* bp = mem_keys + (long)colB * DDIM + khalf;
      v8f acc = {};
#pragma unroll 4
      for (int k0 = 0; k0 < DDIM; k0 += 4) {
        v2f a = *(const v2f*)(ap + k0);             // ds_load_b64
        v2f b = *(const v2f*)(bp + k0);             // global_load_b64 (L2 resident)
        acc = __builtin_amdgcn_wmma_f32_16x16x4_f32(
            false, a, false, b, (short)0, acc, false, false);
      }
      const int m0 = (lane >> 4) * 8;
      const int nn = tile * 16 + (lane & 15);
#pragma unroll
      for (int r = 0; r < 8; ++r)
        sc[(m0 + r) * SC_STRIDE + nn] = acc[r];
    }
    __syncthreads();

    // wave w merges the 512 chunk scores of token w into its top-32 list
#pragma unroll 1
    for (int j = 0; j < NT / 32; ++j) {
      const int   col = lane + 32 * j;
      const float s   = sc[w * SC_STRIDE + col];
      const int   gi  = chunkBase + col;
      bool pend = (s > curMin);
      while (true) {
        unsigned long long bal = __ballot(pend);
        if (bal == 0ull) break;
        const int leader = (int)(__ffsll(bal) - 1);
        if (lane == leader) {
          tv[w * KSEL + minPos] = s;
          ti[w * KSEL + minPos] = gi;
          pend = false;
        }
        // wave-parallel rescan for new min & its position
        float v = tv[w * KSEL + lane];
        int   p = lane;
#pragma unroll
        for (int o = 16; o > 0; o >>= 1) {
          float v2 = __shfl_xor(v, o, 32);
          int   p2 = __shfl_xor(p, o, 32);
          if (v2 < v) { v = v2; p = p2; }
        }
        curMin = v; minPos = p;
        pend = pend && (s > curMin);
      }
    }
    __syncthreads();
  }

  // ---------- Phase 3: softmax over top-32, gather mem_values, fused = x + ctx ----------
  {
    const float myv = tv[w * KSEL + lane];
    const int   myi = ti[w * KSEL + lane];
    const float mx  = wave_max_f(myv);
    const float e   = __expf(myv - mx);
    const float sum = wave_sum_f(e);
    const float wn  = e / sum;

    float acc[8];
#pragma unroll
    for (int q = 0; q < 8; ++q) acc[q] = 0.f;
#pragma unroll 1
    for (int k = 0; k < KSEL; ++k) {
      const float wk = __shfl(wn, k, 32);
      const int   id = __shfl(myi, k, 32);
      const float4* vp = (const float4*)(mem_values + (long)id * DDIM + lane * 8);
      const float4 a0 = vp[0];
      const float4 a1 = vp[1];
      acc[0] += wk * a0.x; acc[1] += wk * a0.y; acc[2] += wk * a0.z; acc[3] += wk * a0.w;
      acc[4] += wk * a1.x; acc[5] += wk * a1.y; acc[6] += wk * a1.z; acc[7] += wk * a1.w;
    }
#pragma unroll
    for (int q = 0; q < 8; ++q) xs[w * DDIM + lane * 8 + q] += acc[q];
  }
  __syncthreads();

  // ---------- Phase 4: h = fused @ W_fuse + b_fuse (WMMA), h tile -> LDS ----------
  {
    const int colB = w * 16 + (lane & 15);
    const float* ap = xs + rowA * DDIM + khalf;
    v8f acc = {};
#pragma unroll 4
    for (int k0 = 0; k0 < DDIM; k0 += 4) {
      v2f a = *(const v2f*)(ap + k0);
      const int kk = k0 + khalf;
      v2f b;
      b.x = W_fuse[(long)kk * DDIM + colB];
      b.y = W_fuse[(long)(kk + 1) * DDIM + colB];
      acc = __builtin_amdgcn_wmma_f32_16x16x4_f32(
          false, a, false, b, (short)0, acc, false, false);
    }
    const float bf = b_fuse[colB];
    const int m0 = (lane >> 4) * 8;
#pragma unroll
    for (int r = 0; r < 8; ++r)
      sc[(m0 + r) * DDIM + colB] = acc[r] + bf;     // h tile, stride DDIM
  }
  __syncthreads();

  // ---------- Phase 5: LayerNorm + ReLU (wave w handles token w's row) ----------
  {
    const int d0 = lane * 8;
    float h[8];
#pragma unroll
    for (int q = 0; q < 8; ++q) h[q] = sc[w * DDIM + d0 + q];
    float s1 = 0.f, s2 = 0.f;
#pragma unroll
    for (int q = 0; q < 8; ++q) { s1 += h[q]; s2 += h[q] * h[q]; }
    s1 = wave_sum_f(s1);
    s2 = wave_sum_f(s2);
    const float mu   = s1 * (1.0f / DDIM);
    const float var  = s2 * (1.0f / DDIM) - mu * mu;
    const float rstd = rsqrtf(var + LN_EPS);
#pragma unroll
    for (int q = 0; q < 8; ++q) {
      const float g = ln_g[d0 + q];
      const float b = ln_b[d0 + q];
      const float v = (h[q] - mu) * rstd * g + b;
      sc[w * DDIM + d0 + q] = fmaxf(v, 0.f);
    }
  }
  __syncthreads();

  // ---------- Phase 6: out = h @ W_op + b_op (WMMA), store to global ----------
  {
    const int colB = w * 16 + (lane & 15);
    const float* ap = sc + rowA * DDIM + khalf;
    v8f acc = {};
#pragma unroll 4
    for (int k0 = 0; k0 < DDIM; k0 += 4) {
      v2f a = *(const v2f*)(ap + k0);
      const int kk = k0 + khalf;
      v2f b;
      b.x = W_op[(long)kk * DDIM + colB];
      b.y = W_op[(long)(kk + 1) * DDIM + colB];
      acc = __builtin_amdgcn_wmma_f32_16x16x4_f32(
          false, a, false, b, (short)0, acc, false, false);
    }
    const float bo = b_op[colB];
    const int m0 = (lane >> 4) * 8;
#pragma unroll
    for (int r = 0; r < 8; ++r)
      out[(tokBase + m0 + r) * DDIM + colB] = acc[r] + bo;
  }
}

extern "C" void kernel_launch(void* const* d_in, const int* in_sizes, int n_in,
                              void* d_out, int out_size, void* d_ws, size_t ws_size,
                              hipStream_t stream) {
  const float* x          = (const float*)d_in[0];
  const float* mem_keys   = (const float*)d_in[1];
  const float* mem_values = (const float*)d_in[2];
  const float* W_fuse     = (const float*)d_in[3];
  const float* b_fuse     = (const float*)d_in[4];
  const float* ln_g       = (const float*)d_in[5];
  const float* ln_b       = (const float*)d_in[6];
  const float* W_op       = (const float*)d_in[7];
  const float* b_op       = (const float*)d_in[8];
  // d_in[9] = top_k scalar (32), compiled in as KSEL
  float* out = (float*)d_out;

  dim3 grid(8192 / TOK_PER_WG);   // B*S / 16 = 512 workgroups
  dim3 block(512);                // 16 waves of 32
  hipLaunchKernelGGL(core_processor_kernel, grid, block, 0, stream,
                     x, mem_keys, mem_values, W_fuse, b_fuse, ln_g, ln_b,
                     W_op, b_op, out);
}